// GraphSAGE_876173328541
// MI455X (gfx1250) — compile-verified
//
#include <hip/hip_runtime.h>
#include <hip/hip_bf16.h>

#define IN_C  128
#define HID_C 128
#define OUT_C 64

typedef float v2f __attribute__((ext_vector_type(2)));
typedef float v8f __attribute__((ext_vector_type(8)));

// ---------------------------------------------------------------------------
// Zero-fill scratch (grid-stride).
// ---------------------------------------------------------------------------
__global__ void zero_f32(float* __restrict__ p, size_t n) {
    size_t i = (size_t)blockIdx.x * blockDim.x + threadIdx.x;
    size_t stride = (size_t)gridDim.x * blockDim.x;
    for (; i < n; i += stride) p[i] = 0.0f;
}

// ---------------------------------------------------------------------------
// Edge scatter: one wave32 per edge. Each lane moves 4 consecutive channels
// (float4 gather from src row, 4x global_atomic_add_f32 into dst row).
// Feature width is 128 for both layers. Lane 0 optionally accumulates degree.
// Roofline: ~1.6 GB of L2-resident traffic per layer -> this kernel is the
// bandwidth-bound bulk of the workload; the f32 atomics resolve in L2.
// ---------------------------------------------------------------------------
__global__ __launch_bounds__(256) void sage_scatter(
    const float* __restrict__ feat,   // [n_nodes, 128] source features
    const int*   __restrict__ srcI,   // [E]
    const int*   __restrict__ dstI,   // [E]
    float*       __restrict__ msg,    // [n_nodes, 128] accumulator (pre-zeroed)
    float*       __restrict__ deg,    // [n_nodes] or nullptr
    int n_edges)
{
    int t = blockIdx.x * blockDim.x + threadIdx.x;
    int e = t >> 5;                 // 32 lanes per edge
    if (e >= n_edges) return;
    int lane = t & 31;
    int s = srcI[e];
    int d = dstI[e];
    const float4 v = *reinterpret_cast<const float4*>(feat + (size_t)s * 128 + lane * 4);
    float* out = msg + (size_t)d * 128 + lane * 4;
    atomicAdd(out + 0, v.x);
    atomicAdd(out + 1, v.y);
    atomicAdd(out + 2, v.z);
    atomicAdd(out + 3, v.w);
    if (deg != nullptr && lane == 0) atomicAdd(deg + d, 1.0f);
}

// ---------------------------------------------------------------------------
// Fused SAGE layer GEMM via fp32 WMMA (V_WMMA_F32_16X16X4_F32).
//   out[n, :] = act( (msg[n,:] / max(deg[n],1)) @ Wl + self[n,:] @ Wr + bias )
// One wave per 16x16 output tile; K-loop in steps of 4, two WMMAs per step.
//
// A-operand (16x4 f32) per-lane layout (ISA 7.12.2):
//   lanes 0-15 : row M = lane,    VGPR0 = K0, VGPR1 = K1
//   lanes 16-31: row M = lane-16, VGPR0 = K2, VGPR1 = K3
// B-operand (4x16 f32): lanes 0-15 hold N=0..15 of K0/K1 (VGPR0/1),
//   lanes 16-31 hold K2/K3. C/D: 8 VGPRs, VGPR v = row (v + 8*(lane>=16)),
//   col = lane%16.
// ---------------------------------------------------------------------------
template <int KDIM, int COUT, bool RELU>
__global__ __launch_bounds__(128) void sage_gemm(
    const float* __restrict__ self_feat,  // [n, KDIM]
    const float* __restrict__ msg,        // [n, KDIM]
    const float* __restrict__ deg,        // [n]
    const float* __restrict__ Wl,         // [KDIM, COUT] row-major
    const float* __restrict__ Wr,         // [KDIM, COUT] row-major
    const float* __restrict__ bias,       // [COUT]
    float*       __restrict__ out,        // [n, COUT]
    int n_nodes)
{
    constexpr int TPR = COUT / 16;            // col tiles per node tile
    const int lane = threadIdx.x;             // 0..31
    const int wave = threadIdx.y;             // 0..3
    const int tile = blockIdx.x * 4 + wave;
    const int node_tile = tile / TPR;
    const int col_tile  = tile % TPR;
    if (node_tile * 16 >= n_nodes) return;    // wave-uniform -> EXEC stays full

    const int r    = lane & 15;
    const int row  = node_tile * 16 + r;      // A row this lane feeds
    const int col  = col_tile * 16 + r;       // B/D column this lane feeds
    const int khl  = (lane >> 4) << 1;        // 0 for lanes 0-15, 2 for 16-31

    const float invd = 1.0f / fmaxf(deg[row], 1.0f);
    const float* srow = self_feat + (size_t)row * KDIM;
    const float* mrow = msg       + (size_t)row * KDIM;

    v8f acc = {};
    for (int k0 = 0; k0 < KDIM; k0 += 4) {
        const int k = k0 + khl;
        v2f a, b;
        // mean-aggregated term: (msg/deg) @ Wl
        a.x = mrow[k]     * invd;
        a.y = mrow[k + 1] * invd;
        b.x = Wl[(size_t)k       * COUT + col];
        b.y = Wl[(size_t)(k + 1) * COUT + col];
        acc = __builtin_amdgcn_wmma_f32_16x16x4_f32(
                  false, a, false, b, (short)0, acc, false, false);
        // self/root term: x @ Wr
        a.x = srow[k];
        a.y = srow[k + 1];
        b.x = Wr[(size_t)k       * COUT + col];
        b.y = Wr[(size_t)(k + 1) * COUT + col];
        acc = __builtin_amdgcn_wmma_f32_16x16x4_f32(
                  false, a, false, b, (short)0, acc, false, false);
    }

    const float bv = bias[col];
    const int mbase = node_tile * 16 + ((lane >> 4) << 3);   // +8 for hi lanes
#pragma unroll
    for (int v = 0; v < 8; ++v) {
        float val = acc[v] + bv;
        if (RELU) val = fmaxf(val, 0.0f);
        out[(size_t)(mbase + v) * COUT + col] = val;
    }
}

// ---------------------------------------------------------------------------
// Launch: zero(msg+deg) -> scatter(x) -> gemm1(relu)->h -> zero(msg)
//         -> scatter(h) -> gemm2 -> d_out
// Workspace layout (floats): msg[N*128] | deg[N] | h[N*128]  (~103 MB)
// ---------------------------------------------------------------------------
extern "C" void kernel_launch(void* const* d_in, const int* in_sizes, int n_in,
                              void* d_out, int out_size, void* d_ws, size_t ws_size,
                              hipStream_t stream) {
    const float* x   = (const float*)d_in[0];
    const int*   ei  = (const int*)  d_in[1];
    const float* W1l = (const float*)d_in[2];
    const float* W1r = (const float*)d_in[3];
    const float* b1  = (const float*)d_in[4];
    const float* W2l = (const float*)d_in[5];
    const float* W2r = (const float*)d_in[6];
    const float* b2  = (const float*)d_in[7];
    float* out = (float*)d_out;

    const int n_nodes = in_sizes[0] / IN_C;   // 100000 (= 6250 * 16, no tail)
    const int n_edges = in_sizes[1] / 2;      // 1.6M
    const int* src = ei;
    const int* dst = ei + n_edges;

    float* msg = (float*)d_ws;                         // [n, 128]
    float* deg = msg + (size_t)n_nodes * HID_C;        // [n]
    float* h   = deg + n_nodes;                        // [n, 128]

    const size_t msg_elems = (size_t)n_nodes * HID_C;

    // Layer 1
    zero_f32<<<2048, 256, 0, stream>>>(msg, msg_elems + n_nodes);   // msg+deg contiguous
    {
        long long threads = (long long)n_edges * 32;
        int blocks = (int)((threads + 255) / 256);
        sage_scatter<<<blocks, 256, 0, stream>>>(x, src, dst, msg, deg, n_edges);
    }
    {
        int tiles  = (n_nodes / 16) * (HID_C / 16);
        int blocks = (tiles + 3) / 4;
        sage_gemm<IN_C, HID_C, true><<<blocks, dim3(32, 4), 0, stream>>>(
            x, msg, deg, W1l, W1r, b1, h, n_nodes);
    }

    // Layer 2 (reuse msg buffer; deg is graph-invariant)
    zero_f32<<<2048, 256, 0, stream>>>(msg, msg_elems);
    {
        long long threads = (long long)n_edges * 32;
        int blocks = (int)((threads + 255) / 256);
        sage_scatter<<<blocks, 256, 0, stream>>>(h, src, dst, msg, nullptr, n_edges);
    }
    {
        int tiles  = (n_nodes / 16) * (OUT_C / 16);
        int blocks = (tiles + 3) / 4;
        sage_gemm<HID_C, OUT_C, false><<<blocks, dim3(32, 4), 0, stream>>>(
            h, msg, deg, W2l, W2r, b2, out, n_nodes);
    }
}